// GraphTransformer_5652176962026
// MI455X (gfx1250) — compile-verified
//
#include <hip/hip_runtime.h>
#include <math.h>

#define N_NODES 50000
#define N_EDGES 800000
#define HIDDIM  128
#define OUT_DIM 90
#define HEADS   4
#define DHEAD   32

#define MTILES  5                 // 16-row tiles per block
#define MROWS   (MTILES * 16)     // 80 rows per block; 50000 = 625 * 80
#define LDS_STRIDE 132            // 128 + 4: 16B-aligned rows, 4-bank rotation per row

typedef __attribute__((ext_vector_type(2))) float v2f;
typedef __attribute__((ext_vector_type(8))) float v8f;

// ---------------- input MLP layer 1: [N,6] @ [6,128] + b, relu ----------------
__global__ void in_gemm6_relu(const float* __restrict__ x, const float* __restrict__ w,
                              const float* __restrict__ b, float* __restrict__ out) {
  int i = blockIdx.x * blockDim.x + threadIdx.x;
  if (i >= N_NODES * HIDDIM) return;
  int row = i >> 7, col = i & 127;
  float s = b[col];
#pragma unroll
  for (int k = 0; k < 6; ++k) s += x[row * 6 + k] * w[k * HIDDIM + col];
  out[i] = fmaxf(s, 0.f);
}

// ---- shared device helper: stage A tile to LDS (all 256 threads participate) ----
__device__ inline void stage_A(const float* __restrict__ A, float* __restrict__ As,
                               int row0) {
  for (int v = threadIdx.x; v < (MROWS * HIDDIM) / 4; v += 256) {
    int elem = v << 2;
    int r = elem >> 7, c = elem & 127;
    float4 src = *(const float4*)(A + (size_t)(row0 + r) * HIDDIM + c);
    *(float4*)(&As[r * LDS_STRIDE + c]) = src;
  }
}

// ---- per-wave 80x16 tile GEMM body: acc over K=128 from LDS A + global W ----
template <int NC>
__device__ inline void gemm_tile(const float* __restrict__ As, const float* __restrict__ W,
                                 int lrow, int khalf, int col, v8f acc[MTILES]) {
  int   ccol  = col;
  float bmask = 1.f;
  if constexpr ((NC & 15) != 0) {        // only the padded [128,90] GEMM pays for this
    ccol  = col < NC ? col : NC - 1;
    bmask = col < NC ? 1.f : 0.f;
  }
#pragma unroll 4
  for (int k = 0; k < HIDDIM; k += 4) {
    int ka = k + khalf * 2;
    v2f bb;
    if constexpr ((NC & 15) != 0) {
      bb.x = W[ka * NC + ccol] * bmask; bb.y = W[(ka + 1) * NC + ccol] * bmask;
    } else {
      bb.x = W[ka * NC + ccol];         bb.y = W[(ka + 1) * NC + ccol];
    }
#pragma unroll
    for (int m = 0; m < MTILES; ++m) {
      float2 av = *(const float2*)(&As[(m * 16 + lrow) * LDS_STRIDE + ka]);
      v2f a; a.x = av.x; a.y = av.y;
      acc[m] = __builtin_amdgcn_wmma_f32_16x16x4_f32(false, a, false, bb, (short)0,
                                                     acc[m], false, false);
    }
  }
}

template <int RELU, int NC>
__device__ inline void store_tile(float* __restrict__ C, const float* __restrict__ bias,
                                  int row0, int khalf, int col, const v8f acc[MTILES]) {
  if ((NC & 15) == 0 || col < NC) {
    float bv = bias[col];
#pragma unroll
    for (int m = 0; m < MTILES; ++m) {
#pragma unroll
      for (int r = 0; r < 8; ++r) {      // VGPR r -> M = r (+8 for lanes 16-31)
        float vv = acc[m][r] + bv;
        if (RELU) vv = fmaxf(vv, 0.f);
        C[(size_t)(row0 + m * 16 + khalf * 8 + r) * NC + col] = vv;
      }
    }
  }
}

// ---------------- single fp32 WMMA GEMM: C = act(A[N,128] @ W[128,NC] + bias) -------
template <int RELU, int NC>
__global__ void wmma_gemm(const float* __restrict__ A, const float* __restrict__ W,
                          const float* __restrict__ bias, float* __restrict__ C) {
  __shared__ float As[MROWS * LDS_STRIDE];
  int row0 = blockIdx.x * MROWS;
  stage_A(A, As, row0);
  __syncthreads();

  int wave = threadIdx.x >> 5, lane = threadIdx.x & 31;
  int col0 = wave << 4;
  if (col0 < NC) {                       // wave-uniform (NC=90: waves 6,7 idle)
    int lrow = lane & 15, khalf = lane >> 4;
    int col  = col0 + lrow;
    v8f acc[MTILES] = {};
    gemm_tile<NC>(As, W, lrow, khalf, col, acc);
    store_tile<RELU, NC>(C, bias, row0, khalf, col, acc);
  }
}

// ---------------- fused Q/K/V/skip: 4 GEMMs sharing one LDS-staged A tile -----------
__global__ void wmma_gemm4(const float* __restrict__ A,
                           const float* __restrict__ W0, const float* __restrict__ b0,
                           float* __restrict__ C0,
                           const float* __restrict__ W1, const float* __restrict__ b1,
                           float* __restrict__ C1,
                           const float* __restrict__ W2, const float* __restrict__ b2,
                           float* __restrict__ C2,
                           const float* __restrict__ W3, const float* __restrict__ b3,
                           float* __restrict__ C3) {
  __shared__ float As[MROWS * LDS_STRIDE];
  int row0 = blockIdx.x * MROWS;
  stage_A(A, As, row0);
  __syncthreads();

  int wave = threadIdx.x >> 5, lane = threadIdx.x & 31;
  int lrow = lane & 15, khalf = lane >> 4;
  int col  = (wave << 4) + lrow;

  const float* Ws[4] = {W0, W1, W2, W3};
  const float* bs[4] = {b0, b1, b2, b3};
  float*       Cs[4] = {C0, C1, C2, C3};
#pragma unroll
  for (int t = 0; t < 4; ++t) {
    v8f acc[MTILES] = {};
    gemm_tile<HIDDIM>(As, Ws[t], lrow, khalf, col, acc);
    store_tile<0, HIDDIM>(Cs[t], bs[t], row0, khalf, col, acc);
  }
}

// ---------------- attention edge phase ----------------
__global__ void init_buffers(float* __restrict__ AGG, float* __restrict__ Mx,
                             float* __restrict__ Sm) {
  int i = blockIdx.x * blockDim.x + threadIdx.x;
  if (i < N_NODES * HIDDIM) AGG[i] = 0.f;
  if (i < N_NODES * HEADS) { Mx[i] = -__builtin_huge_valf(); Sm[i] = 0.f; }
}

__global__ void edge_scores(const float* __restrict__ Q, const float* __restrict__ K,
                            const int* __restrict__ ei, float* __restrict__ alpha) {
  int i = blockIdx.x * blockDim.x + threadIdx.x;
  if (i >= N_EDGES * HEADS) return;
  int e = i >> 2, hh = i & 3;
  int s = ei[e], t = ei[N_EDGES + e];
  const float4* q = (const float4*)(Q + (size_t)t * HIDDIM + hh * DHEAD);
  const float4* k = (const float4*)(K + (size_t)s * HIDDIM + hh * DHEAD);
  float acc = 0.f;
#pragma unroll
  for (int j = 0; j < 8; ++j) {
    float4 a = q[j], b = k[j];
    acc += a.x * b.x + a.y * b.y + a.z * b.z + a.w * b.w;
  }
  alpha[i] = acc * 0.17677669529663687f;   // 1/sqrt(32)
}

__device__ inline void atomicMaxF(float* a, float v) {
  if (v >= 0.f) atomicMax((int*)a, __float_as_int(v));
  else          atomicMin((unsigned int*)a, __float_as_uint(v));
}

__global__ void edge_max(const float* __restrict__ alpha, const int* __restrict__ ei,
                         float* __restrict__ Mx) {
  int i = blockIdx.x * blockDim.x + threadIdx.x;
  if (i >= N_EDGES * HEADS) return;
  int e = i >> 2, hh = i & 3;
  int t = ei[N_EDGES + e];
  atomicMaxF(&Mx[t * HEADS + hh], alpha[i]);
}

__global__ void edge_expsum(float* __restrict__ alpha, const int* __restrict__ ei,
                            const float* __restrict__ Mx, float* __restrict__ Sm) {
  int i = blockIdx.x * blockDim.x + threadIdx.x;
  if (i >= N_EDGES * HEADS) return;
  int e = i >> 2, hh = i & 3;
  int t = ei[N_EDGES + e];
  float ex = expf(alpha[i] - Mx[t * HEADS + hh]);
  alpha[i] = ex;
  atomicAdd(&Sm[t * HEADS + hh], ex);
}

// one thread per (edge, feature) -> single f32 atomic add each
__global__ void edge_agg(const float* __restrict__ alpha, const float* __restrict__ V,
                         const int* __restrict__ ei, const float* __restrict__ Sm,
                         float* __restrict__ AGG) {
  int i = blockIdx.x * blockDim.x + threadIdx.x;
  if (i >= N_EDGES * HIDDIM) return;
  int e = i >> 7, j = i & 127;
  int hh = j >> 5;
  int s = ei[e], t = ei[N_EDGES + e];
  float a = alpha[e * HEADS + hh] / Sm[t * HEADS + hh];
  atomicAdd(&AGG[(size_t)t * HIDDIM + j], a * V[(size_t)s * HIDDIM + j]);
}

// h += relu( head_mean(AGG) broadcast over heads + skip )
__global__ void node_update(float* __restrict__ h, const float* __restrict__ AGG,
                            const float* __restrict__ S) {
  int i = blockIdx.x * blockDim.x + threadIdx.x;
  if (i >= N_NODES * HIDDIM) return;
  int n = i >> 7, j = i & 127, jd = j & 31;
  const float* ag = AGG + (size_t)n * HIDDIM;
  float mean = 0.25f * (ag[jd] + ag[32 + jd] + ag[64 + jd] + ag[96 + jd]);
  h[i] += fmaxf(mean + S[i], 0.f);
}

// ---------------- host-side launch ----------------
extern "C" void kernel_launch(void* const* d_in, const int* in_sizes, int n_in,
                              void* d_out, int out_size, void* d_ws, size_t ws_size,
                              hipStream_t stream) {
  (void)in_sizes; (void)n_in; (void)out_size; (void)ws_size;
  const float* x     = (const float*)d_in[0];
  const int*   ei    = (const int*)d_in[1];
  const float* w_in1 = (const float*)d_in[2];
  const float* b_in1 = (const float*)d_in[3];
  const float* w_in2 = (const float*)d_in[4];
  const float* b_in2 = (const float*)d_in[5];
  const float* wq[2]  = {(const float*)d_in[6],  (const float*)d_in[14]};
  const float* bq[2]  = {(const float*)d_in[7],  (const float*)d_in[15]};
  const float* wk[2]  = {(const float*)d_in[8],  (const float*)d_in[16]};
  const float* bk[2]  = {(const float*)d_in[9],  (const float*)d_in[17]};
  const float* wv[2]  = {(const float*)d_in[10], (const float*)d_in[18]};
  const float* bv[2]  = {(const float*)d_in[11], (const float*)d_in[19]};
  const float* wsk[2] = {(const float*)d_in[12], (const float*)d_in[20]};
  const float* bsk[2] = {(const float*)d_in[13], (const float*)d_in[21]};
  const float* w_out1 = (const float*)d_in[22];
  const float* b_out1 = (const float*)d_in[23];
  const float* w_out2 = (const float*)d_in[24];
  const float* b_out2 = (const float*)d_in[25];

  float* p = (float*)d_ws;
  float* h     = p; p += (size_t)N_NODES * HIDDIM;
  float* Sskip = p; p += (size_t)N_NODES * HIDDIM;
  float* Q     = p; p += (size_t)N_NODES * HIDDIM;
  float* Kx    = p; p += (size_t)N_NODES * HIDDIM;
  float* Vx    = p; p += (size_t)N_NODES * HIDDIM;
  float* AGG   = p; p += (size_t)N_NODES * HIDDIM;
  float* alpha = p; p += (size_t)N_EDGES * HEADS;
  float* Mx    = p; p += (size_t)N_NODES * HEADS;
  float* Sm    = p; p += (size_t)N_NODES * HEADS;

  const int TB = 256;
  dim3 gNH((N_NODES * HIDDIM + TB - 1) / TB);
  dim3 gEH((N_EDGES * HEADS + TB - 1) / TB);
  dim3 gEF((N_EDGES * HIDDIM + TB - 1) / TB);
  dim3 gGemm(N_NODES / MROWS);   // 50000 = 625 * 80, exact

  in_gemm6_relu<<<gNH, TB, 0, stream>>>(x, w_in1, b_in1, Q);
  wmma_gemm<1, HIDDIM><<<gGemm, TB, 0, stream>>>(Q, w_in2, b_in2, h);

  for (int l = 0; l < 2; ++l) {
    wmma_gemm4<<<gGemm, TB, 0, stream>>>(h,
        wsk[l], bsk[l], Sskip,
        wq[l],  bq[l],  Q,
        wk[l],  bk[l],  Kx,
        wv[l],  bv[l],  Vx);
    init_buffers<<<gNH, TB, 0, stream>>>(AGG, Mx, Sm);
    edge_scores<<<gEH, TB, 0, stream>>>(Q, Kx, ei, alpha);
    edge_max   <<<gEH, TB, 0, stream>>>(alpha, ei, Mx);
    edge_expsum<<<gEH, TB, 0, stream>>>(alpha, ei, Mx, Sm);
    edge_agg   <<<gEF, TB, 0, stream>>>(alpha, Vx, ei, Sm, AGG);
    node_update<<<gNH, TB, 0, stream>>>(h, AGG, Sskip);
  }

  wmma_gemm<1, HIDDIM><<<gGemm, TB, 0, stream>>>(h, w_out1, b_out1, Q);
  wmma_gemm<0, OUT_DIM><<<gGemm, TB, 0, stream>>>(Q, w_out2, b_out2, (float*)d_out);
}